// VCM_24146306138323
// MI455X (gfx1250) — compile-verified
//
#include <hip/hip_runtime.h>
#include <hip/hip_bf16.h>
#include <math.h>

// ---------------------------------------------------------------------------
// Problem constants (from the reference)
// ---------------------------------------------------------------------------
#define B_   16
#define C_   512
#define HW_  40
#define N_   1600          // HW_*HW_
#define P_   256

// ---------------------------------------------------------------------------
// WMMA f32 fragment types (gfx1250, wave32)
// ---------------------------------------------------------------------------
typedef __attribute__((ext_vector_type(2))) float v2f;   // A/B frag of 16x16x4 f32
typedef __attribute__((ext_vector_type(8))) float v8f;   // C/D frag (16x16 f32)

// ---------------------------------------------------------------------------
// Tiled WMMA GEMM:  C[M,N] (+)= A[M,K] * B'[K,N]
//   A[m,k] = Ap[m*a_sm + k*a_sk]            (stride-parameterized -> transposes free)
//   B'[k,n] depends on BMODE:
//     PLAIN   : Bp[k*b_sk + n*b_sn]
//     SOFTMAX : exp(Bp[k*b_sk + n*b_sn] - smax[k]) * sinv[k]   (softmax over k fused)
//     CONV    : zero-padded spatial shift (implicit 3x3-conv tap)
// Block tile 64x128, BK=16. 128 threads = 4 waves (2x2); wave tile 32x64 =
// 2x4 V_WMMA_F32_16X16X4_F32 fragments (8 v8f accumulators). Register-staged
// double buffering overlaps HBM loads of k-block t+1 with WMMAs on block t.
// ---------------------------------------------------------------------------
#define BM 64
#define BN 128
#define BK 16
#define GTHREADS 128       // GEMM workgroup size
#define STHREADS 256       // stats workgroup size
#define APAD 8             // 2*(BM+APAD) % 64 == 16 -> half-waves hit disjoint banks
#define BPAD 8

constexpr int BMODE_PLAIN   = 0;
constexpr int BMODE_SOFTMAX = 1;
constexpr int BMODE_CONV    = 2;

#define AREG 8             // BK*BM / GTHREADS
#define BREG 16            // BK*BN / GTHREADS

template <int BMODE, bool KFAST>
__device__ __forceinline__ void stage_tiles(
    const float* __restrict__ Ap, long a_sm, long a_sk,
    const float* __restrict__ Bp, long b_sk, long b_sn,
    const float* __restrict__ smax, const float* __restrict__ sinv,
    int m0, int n0, int k0, int M, int N, int K, int dy, int dx, int tid,
    float (&areg)[AREG], float (&breg)[BREG])
{
    // A: element j -> As[(tid>>6) + 2j][tid&63]
    const int amm = tid & (BM - 1);
    const int akb = tid >> 6;
#pragma unroll
    for (int j = 0; j < AREG; ++j) {
        const int gm = m0 + amm, gk = k0 + akb + 2 * j;
        float v = 0.f;
        if (gm < M && gk < K) v = Ap[(long)gm * a_sm + (long)gk * a_sk];
        areg[j] = v;
    }
    // B: n-fast: element j -> Bs[j][tid]; k-fast: -> Bs[tid&15][(tid>>4)+8j]
#pragma unroll
    for (int j = 0; j < BREG; ++j) {
        int kk, nn;
        if (KFAST) { kk = tid & (BK - 1); nn = (tid >> 4) + 8 * j; }
        else       { nn = tid;            kk = j; }
        const int gk = k0 + kk, gn = n0 + nn;
        float v = 0.f;
        if (BMODE == BMODE_CONV) {
            if (gn < N && gk < K) {
                const int y  = gn / HW_;
                const int x  = gn - y * HW_;
                const int yy = y + dy, xx = x + dx;
                if (yy >= 0 && yy < HW_ && xx >= 0 && xx < HW_)
                    v = Bp[(long)gk * b_sk + (long)(yy * HW_ + xx) * b_sn];
            }
        } else if (BMODE == BMODE_SOFTMAX) {
            if (gn < N && gk < K)
                v = __expf(Bp[(long)gk * b_sk + (long)gn * b_sn] - smax[gk]) * sinv[gk];
        } else {
            if (gn < N && gk < K)
                v = Bp[(long)gk * b_sk + (long)gn * b_sn];
        }
        breg[j] = v;
    }
}

template <int BMODE, bool KFAST, bool ACCUM>
__global__ __launch_bounds__(GTHREADS)
void wmma_gemm_kernel(const float* __restrict__ Ap, long a_sm, long a_sk, long a_batch,
                      const float* __restrict__ Bp, long b_sk, long b_sn, long b_batch,
                      float* __restrict__ Cp, int ldc, long c_batch,
                      int M, int N, int K,
                      const float* __restrict__ smax, const float* __restrict__ sinv,
                      int dy, int dx)
{
    const int b = blockIdx.z;
    Ap += (long)b * a_batch;
    Bp += (long)b * b_batch;
    Cp += (long)b * c_batch;
    if (BMODE == BMODE_SOFTMAX) { smax += (long)b * N_; sinv += (long)b * N_; }

    __shared__ float As[BK][BM + APAD];   // [k][m]
    __shared__ float Bs[BK][BN + BPAD];   // [k][n]

    const int tid  = threadIdx.x;
    const int lane = tid & 31;
    const int wave = tid >> 5;            // 4 waves: 2(M) x 2(N)
    const int wm0  = (wave >> 1) * 32;
    const int wn0  = (wave & 1) * 64;
    const int m0   = blockIdx.y * BM;
    const int n0   = blockIdx.x * BN;

    const int fl = lane & 15;             // fragment row/col within 16
    const int kh = (lane >> 4) << 1;      // half-wave K offset: 0 or 2

    v8f acc[2][4];
    {
        v8f z = {};
#pragma unroll
        for (int i = 0; i < 2; ++i)
#pragma unroll
            for (int j = 0; j < 4; ++j) acc[i][j] = z;
    }

    float areg[AREG], breg[BREG];
    stage_tiles<BMODE, KFAST>(Ap, a_sm, a_sk, Bp, b_sk, b_sn, smax, sinv,
                              m0, n0, 0, M, N, K, dy, dx, tid, areg, breg);

    const int amm = tid & (BM - 1);
    const int akb = tid >> 6;

    for (int k0 = 0; k0 < K; k0 += BK) {
        // ---- commit staged registers to LDS
#pragma unroll
        for (int j = 0; j < AREG; ++j) As[akb + 2 * j][amm] = areg[j];
#pragma unroll
        for (int j = 0; j < BREG; ++j) {
            if (KFAST) Bs[tid & (BK - 1)][(tid >> 4) + 8 * j] = breg[j];
            else       Bs[j][tid] = breg[j];
        }
        __syncthreads();

        // ---- prefetch next k-block from HBM while WMMAs run on this one
        if (k0 + BK < K)
            stage_tiles<BMODE, KFAST>(Ap, a_sm, a_sk, Bp, b_sk, b_sn, smax, sinv,
                                      m0, n0, k0 + BK, M, N, K, dy, dx, tid,
                                      areg, breg);

        // ---- compute: 4 k-steps x (2x4) WMMA; EXEC all-ones (uniform flow)
#pragma unroll
        for (int kk = 0; kk < BK; kk += 4) {
            v2f afrag[2], bfrag[4];
#pragma unroll
            for (int t = 0; t < 2; ++t) {
                // A 16x4: lanes 0-15 -> K=kk+0/kk+1, lanes 16-31 -> K=kk+2/kk+3
                afrag[t].x = As[kk + kh + 0][wm0 + t * 16 + fl];
                afrag[t].y = As[kk + kh + 1][wm0 + t * 16 + fl];
            }
#pragma unroll
            for (int t = 0; t < 4; ++t) {
                bfrag[t].x = Bs[kk + kh + 0][wn0 + t * 16 + fl];
                bfrag[t].y = Bs[kk + kh + 1][wn0 + t * 16 + fl];
            }
#pragma unroll
            for (int i = 0; i < 2; ++i)
#pragma unroll
                for (int j = 0; j < 4; ++j)
                    acc[i][j] = __builtin_amdgcn_wmma_f32_16x16x4_f32(
                        false, afrag[i], false, bfrag[j],
                        (short)0, acc[i][j], false, false);
        }
        __syncthreads();
    }

    // ---- store: C/D layout is VGPR r -> M = r + 8*(lane>=16), N = lane&15
#pragma unroll
    for (int i = 0; i < 2; ++i) {
#pragma unroll
        for (int j = 0; j < 4; ++j) {
#pragma unroll
            for (int r = 0; r < 8; ++r) {
                const int m = m0 + wm0 + i * 16 + r + ((lane >> 4) << 3);
                const int n = n0 + wn0 + j * 16 + fl;
                if (m < M && n < N) {
                    float v = acc[i][j][r];
                    if (ACCUM) v += Cp[(long)m * ldc + n];
                    Cp[(long)m * ldc + n] = v;
                }
            }
        }
    }
}

// ---------------------------------------------------------------------------
// Softmax statistics over rows of A[e,q] (normalize over q, for sm_q).
// ---------------------------------------------------------------------------
__global__ __launch_bounds__(STHREADS)
void rowstats_kernel(const float* __restrict__ Aaff,
                     float* __restrict__ rmax, float* __restrict__ rinv)
{
    const int e = blockIdx.x, b = blockIdx.y;
    const float* row = Aaff + ((long)b * N_ + e) * (long)N_;
    __shared__ float red[STHREADS];
    const int tid = threadIdx.x;

    float m = -__builtin_inff();
    for (int q = tid; q < N_; q += STHREADS) m = fmaxf(m, row[q]);
    red[tid] = m; __syncthreads();
    for (int s = STHREADS / 2; s > 0; s >>= 1) {
        if (tid < s) red[tid] = fmaxf(red[tid], red[tid + s]);
        __syncthreads();
    }
    const float mx = red[0];
    __syncthreads();

    float sum = 0.f;
    for (int q = tid; q < N_; q += STHREADS) sum += __expf(row[q] - mx);
    red[tid] = sum; __syncthreads();
    for (int s = STHREADS / 2; s > 0; s >>= 1) {
        if (tid < s) red[tid] += red[tid + s];
        __syncthreads();
    }
    if (tid == 0) {
        rmax[(long)b * N_ + e] = mx;
        rinv[(long)b * N_ + e] = 1.f / red[0];
    }
}

// ---------------------------------------------------------------------------
// Softmax statistics over columns of A[e,q] (normalize over e, for sm_e):
// one thread per column, online max/sum; fully coalesced over q.
// ---------------------------------------------------------------------------
__global__ __launch_bounds__(STHREADS)
void colstats_kernel(const float* __restrict__ Aaff,
                     float* __restrict__ cmax, float* __restrict__ cinv)
{
    const int q = blockIdx.x * STHREADS + threadIdx.x;
    const int b = blockIdx.y;
    if (q >= N_) return;
    const float* Ab = Aaff + (long)b * N_ * N_;
    float m = -__builtin_inff(), s = 0.f;
    for (int e = 0; e < N_; ++e) {
        const float x  = Ab[(long)e * N_ + q];
        const float mn = fmaxf(m, x);
        s = s * __expf(m - mn) + __expf(x - mn);
        m = mn;
    }
    cmax[(long)b * N_ + q] = m;
    cinv[(long)b * N_ + q] = 1.f / s;
}

// ---------------------------------------------------------------------------
// Host orchestration.
// Workspace layout (floats), total ~67.3M floats (~269 MB):
//   [EC | QC] -> reused as ATT1 after affinity, AAF, ATT2, row/col stats.
// ---------------------------------------------------------------------------
extern "C" void kernel_launch(void* const* d_in, const int* in_sizes, int n_in,
                              void* d_out, int out_size, void* d_ws, size_t ws_size,
                              hipStream_t stream)
{
    (void)in_sizes; (void)n_in; (void)out_size; (void)ws_size;

    const float* exemplar = (const float*)d_in[0];  // [B,C,H,W]
    const float* query    = (const float*)d_in[1];  // [B,C,H,W]
    const float* We       = (const float*)d_in[2];  // [P,C]
    const float* Wq       = (const float*)d_in[3];  // [P,C]
    const float* W1       = (const float*)d_in[4];  // [C,C,3,3]
    const float* W2       = (const float*)d_in[5];  // [C,C,3,3]

    float* ws = (float*)d_ws;
    const long EC   = 0;                               // B*P*N
    const long QC   = EC + (long)B_ * P_ * N_;         // B*P*N
    const long ATT1 = 0;                               // B*C*N (overlaps EC+QC, dead by then)
    const long AAF  = QC + (long)B_ * P_ * N_;         // B*N*N
    const long ATT2 = AAF + (long)B_ * N_ * N_;        // B*C*N
    const long RMAX = ATT2 + (long)B_ * C_ * N_;
    const long RINV = RMAX + (long)B_ * N_;
    const long CMAX = RINV + (long)B_ * N_;
    const long CINV = CMAX + (long)B_ * N_;

    float* out1 = (float*)d_out;                       // conv(exemplar_att, W1)
    float* out2 = out1 + (long)B_ * C_ * N_;           // conv(query_att,    W2)

    const dim3 blk(GTHREADS);
    const dim3 sblk(STHREADS);
    const int  gx = (N_ + BN - 1) / BN;                // 13 N-tiles

    // 1) 1x1 projections: ec = We * ex_flat, qc = Wq * q_flat   (M=256,K=512,N=1600)
    {
        dim3 grd(gx, (P_ + BM - 1) / BM, B_);
        wmma_gemm_kernel<BMODE_PLAIN, false, false><<<grd, blk, 0, stream>>>(
            We, C_, 1, 0,   exemplar, N_, 1, (long)C_ * N_,
            ws + EC, N_, (long)P_ * N_,  P_, N_, C_, nullptr, nullptr, 0, 0);
        wmma_gemm_kernel<BMODE_PLAIN, false, false><<<grd, blk, 0, stream>>>(
            Wq, C_, 1, 0,   query,    N_, 1, (long)C_ * N_,
            ws + QC, N_, (long)P_ * N_,  P_, N_, C_, nullptr, nullptr, 0, 0);
    }

    // 2) affinity A[e,q] = sum_p ec[p,e]*qc[p,q]  (M=N=1600, K=256; A read transposed)
    {
        dim3 grd(gx, (N_ + BM - 1) / BM, B_);
        wmma_gemm_kernel<BMODE_PLAIN, false, false><<<grd, blk, 0, stream>>>(
            ws + EC, 1, N_, (long)P_ * N_,   ws + QC, N_, 1, (long)P_ * N_,
            ws + AAF, N_, (long)N_ * N_,  N_, N_, P_, nullptr, nullptr, 0, 0);
    }

    // 3) softmax statistics (row = over q for sm_q; col = over e for sm_e)
    rowstats_kernel<<<dim3(N_, B_), sblk, 0, stream>>>(ws + AAF, ws + RMAX, ws + RINV);
    colstats_kernel<<<dim3((N_ + STHREADS - 1) / STHREADS, B_), sblk, 0, stream>>>(
        ws + AAF, ws + CMAX, ws + CINV);

    // 4) attention products (M=512, K=1600, N=1600), softmax fused into B-load
    {
        dim3 grd(gx, (C_ + BM - 1) / BM, B_);
        // exemplar_att[c,q] = sum_e qf[c,e] * sm_q[e,q]  B[k=e,n=q]=A[e*N+q], stats by e
        wmma_gemm_kernel<BMODE_SOFTMAX, false, false><<<grd, blk, 0, stream>>>(
            query, N_, 1, (long)C_ * N_,   ws + AAF, N_, 1, (long)N_ * N_,
            ws + ATT1, N_, (long)C_ * N_,  C_, N_, N_, ws + RMAX, ws + RINV, 0, 0);
        // query_att[c,e] = sum_q ef[c,q] * sm_e[e,q]     B[k=q,n=e]=A[e*N+q], stats by q
        wmma_gemm_kernel<BMODE_SOFTMAX, true, false><<<grd, blk, 0, stream>>>(
            exemplar, N_, 1, (long)C_ * N_,   ws + AAF, 1, N_, (long)N_ * N_,
            ws + ATT2, N_, (long)C_ * N_,  C_, N_, N_, ws + CMAX, ws + CINV, 0, 0);
    }

    // 5) 3x3 SAME convs as 9 shifted-tap implicit GEMMs (M=512, K=512 per tap, N=1600)
    {
        dim3 grd(gx, (C_ + BM - 1) / BM, B_);
        for (int t = 0; t < 9; ++t) {
            const int dy = t / 3 - 1, dx = t % 3 - 1;
            if (t == 0) {
                wmma_gemm_kernel<BMODE_CONV, false, false><<<grd, blk, 0, stream>>>(
                    W1 + t, (long)C_ * 9, 9, 0,   ws + ATT1, N_, 1, (long)C_ * N_,
                    out1, N_, (long)C_ * N_,  C_, N_, C_, nullptr, nullptr, dy, dx);
                wmma_gemm_kernel<BMODE_CONV, false, false><<<grd, blk, 0, stream>>>(
                    W2 + t, (long)C_ * 9, 9, 0,   ws + ATT2, N_, 1, (long)C_ * N_,
                    out2, N_, (long)C_ * N_,  C_, N_, C_, nullptr, nullptr, dy, dx);
            } else {
                wmma_gemm_kernel<BMODE_CONV, false, true><<<grd, blk, 0, stream>>>(
                    W1 + t, (long)C_ * 9, 9, 0,   ws + ATT1, N_, 1, (long)C_ * N_,
                    out1, N_, (long)C_ * N_,  C_, N_, C_, nullptr, nullptr, dy, dx);
                wmma_gemm_kernel<BMODE_CONV, false, true><<<grd, blk, 0, stream>>>(
                    W2 + t, (long)C_ * 9, 9, 0,   ws + ATT2, N_, 1, (long)C_ * N_,
                    out2, N_, (long)C_ * N_,  C_, N_, C_, nullptr, nullptr, dy, dx);
            }
        }
    }
}